// ContrastiveLearning_41772851921216
// MI455X (gfx1250) — compile-verified
//
#include <hip/hip_runtime.h>
#include <hip/hip_bf16.h>
#include <math.h>

// ---------------------------------------------------------------------------
// ContrastiveLearning on MI455X (gfx1250, wave32).
// N=2048 rows, H=768, D=128. All matrix math via V_WMMA_F32_16X16X4_F32
// (fp32 tensor path -- reference is fp32 and feeds exp/log2).
// Pair kernel: i-tile staged in LDS (shared by 8 waves/block), j-side via
// VMEM; muinv / mu^2 synthesized in-register instead of streamed.
// ---------------------------------------------------------------------------

typedef float v2f __attribute__((ext_vector_type(2)));
typedef float v8f __attribute__((ext_vector_type(8)));

#define N_ROWS 2048
#define HDIM   768
#define DDIM   128

__device__ __forceinline__ v8f wmma_f32_k4(v2f a, v2f b, v8f c) {
  // D = A(16x4,f32) x B(4x16,f32) + C(16x16,f32)
  return __builtin_amdgcn_wmma_f32_16x16x4_f32(
      /*neg_a=*/false, a, /*neg_b=*/false, b,
      /*c_mod=*/(short)0, c, /*reuse_a=*/false, /*reuse_b=*/false);
}

// ---------------------------------------------------------------------------
// Kernel 1: projection GEMM.  h = relu(X);  mu = h@Wm + bm;
// sigma = elu(h@Ws + bs) + 1 + 1e-14.  One wave per 16x16 output tile of the
// combined [768 x 256] weight (col-tiles 0..7 -> mu, 8..15 -> sigma).
// ---------------------------------------------------------------------------
__global__ void proj_kernel(const float* __restrict__ X,
                            const float* __restrict__ Wm, const float* __restrict__ bm,
                            const float* __restrict__ Ws, const float* __restrict__ bs,
                            float* __restrict__ mu, float* __restrict__ sigma) {
  const int wave = (blockIdx.x * blockDim.x + threadIdx.x) >> 5;   // 0..2047
  const int lane = threadIdx.x & 31;
  const int half = lane >> 4;
  const int mrow = lane & 15;
  const int itile = wave >> 4;    // 0..127
  const int ctile = wave & 15;    // 0..15
  const int i0 = itile * 16;
  const bool is_mu = (ctile < 8);
  const int ccol = ((ctile & 7) * 16) + mrow;          // 0..127 inside W
  const float* __restrict__ W    = is_mu ? Wm : Ws;
  const float* __restrict__ bias = is_mu ? bm : bs;

  const float* arow = X + (size_t)(i0 + mrow) * HDIM + half * 2;

  v8f acc = {};
  for (int k = 0; k < HDIM; k += 4) {
    v2f a = *(const v2f*)(arow + k);
    a.x = fmaxf(a.x, 0.f);                 // fused ReLU
    a.y = fmaxf(a.y, 0.f);
    const int kk = k + half * 2;
    v2f b;
    b.x = W[(size_t)kk * DDIM + ccol];
    b.y = W[(size_t)(kk + 1) * DDIM + ccol];
    acc = wmma_f32_k4(a, b, acc);
  }

  const float bv = bias[ccol];
#pragma unroll
  for (int v = 0; v < 8; ++v) {
    const int row = i0 + v + 8 * half;
    const float val = acc[v] + bv;
    if (is_mu) {
      mu[(size_t)row * DDIM + ccol] = val;
    } else {
      // elu(x)+1+1e-14:  x>0 -> x+1+1e-14 ;  x<=0 -> exp(x)+1e-14
      const float s = (val > 0.f) ? (val + 1.f + 1e-14f) : (__expf(val) + 1e-14f);
      sigma[(size_t)row * DDIM + ccol] = s;
    }
  }
}

// ---------------------------------------------------------------------------
// Kernel 2: per-row derived data. One block (128 thr) per row.
//   inv = 1/sigma
//   S_log = sum log sigma, q = sum mu^2*inv, mu2sum = sum mu^2
//   validf = (mask==1 && label>=0); also zero the pair accumulators.
// ---------------------------------------------------------------------------
__global__ void rowstats_kernel(const float* __restrict__ mu, const float* __restrict__ sigma,
                                const int* __restrict__ type_ids, const int* __restrict__ mask,
                                float* __restrict__ inv,
                                float* __restrict__ S_log, float* __restrict__ q,
                                float* __restrict__ mu2sum, float* __restrict__ validf,
                                float* __restrict__ lossnum, float* __restrict__ lossden,
                                float* __restrict__ cnts) {
  __shared__ float sh0[128], sh1[128], sh2[128];
  const int i = blockIdx.x;
  const int d = threadIdx.x;
  const size_t idx = (size_t)i * DDIM + d;
  const float s  = sigma[idx];
  const float m  = mu[idx];
  const float iv = 1.f / s;
  inv[idx] = iv;
  sh0[d] = __logf(s);
  sh1[d] = m * m * iv;
  sh2[d] = m * m;
  __syncthreads();
  for (int off = 64; off > 0; off >>= 1) {
    if (d < off) { sh0[d] += sh0[d + off]; sh1[d] += sh1[d + off]; sh2[d] += sh2[d + off]; }
    __syncthreads();
  }
  if (d == 0) {
    S_log[i]  = sh0[0];
    q[i]      = sh1[0];
    mu2sum[i] = sh2[0];
    validf[i] = (mask[i] == 1 && type_ids[i] >= 0) ? 1.f : 0.f;
    lossnum[i] = 0.f; lossden[i] = 0.f; cnts[i] = 0.f;   // accumulator init
  }
}

// ---------------------------------------------------------------------------
// Kernel 3: fused pairwise KL + NT-Xent partial sums.
// Block = 8 waves sharing one i-tile (staged in LDS: inv/mu/sigma, 24 KB),
// wave w owns j-tile (jgroup*8 + w).  7 fp32 WMMAs per K-step:
//   aF = inv_i . sigma_j      aR = sigma_i . inv_j
//   cF = (mu*inv)_i . mu_j    cR = mu_i . (mu*inv)_j
//   qF = inv_i . mu2_j        qR = mu2_i . inv_j
//   dT = mu_i . mu_j          (duplicate-row mask)
// muinv / mu2 operands are synthesized with v_mul from the staged mu/inv.
// Epilogue: symmetrized KL, exp, masks, 16-lane shuffle reduce over j,
// atomicAdd into per-row num/den/cnt.  The reference 'alldiff' mask exists
// to kill the diagonal & duplicate rows; realized as (i!=j) &&
// ||mu_i-mu_j||^2 > eps*scale (equivalent for continuous data).
// ---------------------------------------------------------------------------
__global__ void pair_kernel(const float* __restrict__ inv, const float* __restrict__ mu,
                            const float* __restrict__ sigma,
                            const float* __restrict__ S_log, const float* __restrict__ q,
                            const float* __restrict__ mu2sum, const float* __restrict__ validf,
                            const int* __restrict__ labels,
                            float* __restrict__ lossnum, float* __restrict__ lossden,
                            float* __restrict__ cnts) {
  __shared__ float s_inv[16 * DDIM], s_mu[16 * DDIM], s_sg[16 * DDIM];

  const int ti = blockIdx.x >> 4;            // 0..127 : i-tile
  const int jg = blockIdx.x & 15;            // j-group
  const int w  = threadIdx.x >> 5;           // wave in block 0..7
  const int tj = jg * 8 + w;                 // 0..127 : j-tile
  const int lane = threadIdx.x & 31;
  const int half = lane >> 4;
  const int mrow = lane & 15;
  const int i0 = ti * 16, j0 = tj * 16;

  // Stage the i-tile (16 contiguous rows = 2048 floats per array) into LDS.
  {
    const float4* gi = (const float4*)(inv   + (size_t)i0 * DDIM);
    const float4* gm = (const float4*)(mu    + (size_t)i0 * DDIM);
    const float4* gs = (const float4*)(sigma + (size_t)i0 * DDIM);
    for (int t = threadIdx.x; t < (16 * DDIM) / 4; t += 256) {
      ((float4*)s_inv)[t] = gi[t];
      ((float4*)s_mu)[t]  = gm[t];
      ((float4*)s_sg)[t]  = gs[t];
    }
  }
  __syncthreads();

  const int abase = mrow * DDIM + half * 2;                  // LDS frag base (i)
  const size_t rj = (size_t)(j0 + mrow) * DDIM + half * 2;   // VMEM frag base (j)

  v8f aF = {}, cF = {}, qF = {}, aR = {}, cR = {}, qR = {}, dT = {};
  for (int k = 0; k < DDIM; k += 4) {
    const v2f inv_i = *(const v2f*)(s_inv + abase + k);      // ds_load_b64
    const v2f mu_i  = *(const v2f*)(s_mu  + abase + k);
    const v2f sg_i  = *(const v2f*)(s_sg  + abase + k);
    const v2f inv_j = *(const v2f*)(inv   + rj + k);         // global_load_b64
    const v2f mu_j  = *(const v2f*)(mu    + rj + k);
    const v2f sg_j  = *(const v2f*)(sigma + rj + k);
    v2f miv_i, m2_i, miv_j, m2_j;                            // in-register
    miv_i.x = mu_i.x * inv_i.x;  miv_i.y = mu_i.y * inv_i.y;
    m2_i.x  = mu_i.x * mu_i.x;   m2_i.y  = mu_i.y * mu_i.y;
    miv_j.x = mu_j.x * inv_j.x;  miv_j.y = mu_j.y * inv_j.y;
    m2_j.x  = mu_j.x * mu_j.x;   m2_j.y  = mu_j.y * mu_j.y;
    aF = wmma_f32_k4(inv_i, sg_j, aF);
    cF = wmma_f32_k4(miv_i, mu_j, cF);
    qF = wmma_f32_k4(inv_i, m2_j, qF);
    aR = wmma_f32_k4(sg_i, inv_j, aR);
    cR = wmma_f32_k4(mu_i, miv_j, cR);
    qR = wmma_f32_k4(m2_i, inv_j, qR);
    dT = wmma_f32_k4(mu_i, mu_j, dT);
  }

  const int j = j0 + mrow;
  const float slj = S_log[j], qj = q[j], m2j = mu2sum[j], vfj = validf[j];
  const int labj = labels[j];

#pragma unroll
  for (int v = 0; v < 8; ++v) {
    const int i = i0 + v + 8 * half;
    const float sli = S_log[i], qi = q[i], m2i = mu2sum[i], vfi = validf[i];
    const int labi = labels[i];

    const float ld  = slj - sli;                                  // sum log(sig_j/sig_i)
    const float ijk = 0.5f * (aF[v] + (qi - 2.f * cF[v] + qF[v]) - (float)DDIM - ld);
    const float jik = 0.5f * (aR[v] + (qj - 2.f * cR[v] + qR[v]) - (float)DDIM + ld);
    const float loss = -0.5f * (ijk + jik);
    const float e = __expf(loss);

    const float diffsq = m2i + m2j - 2.f * dT[v];                 // ||mu_i - mu_j||^2
    const bool dup = (i == j) || (fabsf(diffsq) <= 1e-10f * (m2i + m2j));
    const float keep = dup ? 0.f : 1.f;
    const float denw = keep * vfi * vfj;                          // loss_mask
    const float numw = (labi == labj) ? denw : 0.f;               // loss_weights

    float pn = e * numw, pd = e * denw, pc = numw;
    // reduce across the 16 lanes of this half-wave (same row i, all j in tile)
    for (int mk = 8; mk >= 1; mk >>= 1) {
      pn += __shfl_xor(pn, mk);
      pd += __shfl_xor(pd, mk);
      pc += __shfl_xor(pc, mk);
    }
    if (mrow == 0) {
      atomicAdd(&lossnum[i], pn);
      atomicAdd(&lossden[i], pd);
      atomicAdd(&cnts[i], pc);
    }
  }
}

// ---------------------------------------------------------------------------
// Kernel 4: final NT-Xent reduction over rows -> scalar.
// ---------------------------------------------------------------------------
__global__ void finalize_kernel(const float* __restrict__ lossnum,
                                const float* __restrict__ lossden,
                                const float* __restrict__ cnts,
                                float* __restrict__ out) {
  __shared__ float ssum[256], scnt[256];
  const int t = threadIdx.x;
  float s = 0.f, c = 0.f;
  for (int i = t; i < N_ROWS; i += 256) {
    const float cn = cnts[i];
    if (cn > 0.f) {
      s += -__log2f(lossnum[i]) + __log2f(lossden[i]) + __log2f(cn);
      c += 1.f;
    }
  }
  ssum[t] = s; scnt[t] = c;
  __syncthreads();
  for (int off = 128; off > 0; off >>= 1) {
    if (t < off) { ssum[t] += ssum[t + off]; scnt[t] += scnt[t + off]; }
    __syncthreads();
  }
  if (t == 0) out[0] = ssum[0] / fmaxf(scnt[0], 1.f);
}

// ---------------------------------------------------------------------------
extern "C" void kernel_launch(void* const* d_in, const int* in_sizes, int n_in,
                              void* d_out, int out_size, void* d_ws, size_t ws_size,
                              hipStream_t stream) {
  const float* X  = (const float*)d_in[0];   // ent_embeddings [4,512,768]
  const int*   ty = (const int*)  d_in[1];   // ent_type_ids   [4,512]
  const int*   mk = (const int*)  d_in[2];   // ent_mask       [4,512]
  const float* Wm = (const float*)d_in[3];   // W_mu  [768,128]
  const float* bm = (const float*)d_in[4];   // b_mu  [128]
  const float* Ws = (const float*)d_in[5];   // W_sigma [768,128]
  const float* bs = (const float*)d_in[6];   // b_sigma [128]
  float* out = (float*)d_out;

  // workspace layout (floats)
  float* ws = (float*)d_ws;
  const size_t ND = (size_t)N_ROWS * DDIM;
  float* mu     = ws;
  float* sigma  = ws + 1 * ND;
  float* inv    = ws + 2 * ND;
  float* S_log  = ws + 3 * ND;
  float* q      = S_log + N_ROWS;
  float* m2sum  = S_log + 2 * N_ROWS;
  float* vf     = S_log + 3 * N_ROWS;
  float* lnum   = S_log + 4 * N_ROWS;
  float* lden   = S_log + 5 * N_ROWS;
  float* cnt    = S_log + 6 * N_ROWS;

  // 1) projection: 2048 waves = 128 i-tiles x 16 col-tiles; 8 waves/block
  proj_kernel<<<256, 256, 0, stream>>>(X, Wm, bm, Ws, bs, mu, sigma);
  // 2) per-row stats + accumulator init
  rowstats_kernel<<<N_ROWS, 128, 0, stream>>>(mu, sigma, ty, mk, inv,
                                              S_log, q, m2sum, vf,
                                              lnum, lden, cnt);
  // 3) pairwise: 128 i-tiles x 16 j-groups; 8 waves/block (one j-tile each)
  pair_kernel<<<2048, 256, 0, stream>>>(inv, mu, sigma,
                                        S_log, q, m2sum, vf, ty,
                                        lnum, lden, cnt);
  // 4) scalar reduction
  finalize_kernel<<<1, 256, 0, stream>>>(lnum, lden, cnt, out);
}